// CustomLoss_Z_25031069401263
// MI455X (gfx1250) — compile-verified
//
#include <hip/hip_runtime.h>
#include <stdint.h>

// Problem geometry (fixed by the reference)
#define D_DIM 128
#define H_DIM 192
#define W_DIM 193
#define WD    192                     // width of d = dz
#define TZ    4
#define TY    8
#define NBLK  ((D_DIM/TZ)*(H_DIM/TY)) // 768 stencil workgroups
#define NTOT  (D_DIM*H_DIM*WD)        // 4718592 elements of d
#define NZ0   (D_DIM*H_DIM)           // 24576 elements of z0
#define KRANK ((NZ0-1)/2)             // 12287 lower-median rank
#define FBIG  3.402823466e38f

typedef uint32_t u32;
typedef uint32_t u32x4 __attribute__((ext_vector_type(4)));
typedef uint32_t u32x8 __attribute__((ext_vector_type(8)));

// Low 32 bits of a generic pointer to __shared__ = LDS byte offset.
__device__ __forceinline__ u32 lds_off(const void* p) {
  return (u32)(uintptr_t)p;
}

// CDNA5 async global->LDS copies (ASYNCcnt-tracked). GV mode: 64-bit vaddr, saddr=off.
__device__ __forceinline__ void async_copy_b128(u32 ldsoff, const void* g) {
  asm volatile("global_load_async_to_lds_b128 %0, %1, off"
               :: "v"(ldsoff), "v"((uint64_t)(uintptr_t)g) : "memory");
}
__device__ __forceinline__ void async_copy_b32(u32 ldsoff, const void* g) {
  asm volatile("global_load_async_to_lds_b32 %0, %1, off"
               :: "v"(ldsoff), "v"((uint64_t)(uintptr_t)g) : "memory");
}
__device__ __forceinline__ void wait_async0() {
  asm volatile("s_wait_asynccnt 0x0" ::: "memory");
}

// reflect-pad index (jnp.pad mode='reflect', pad=1)
__device__ __forceinline__ int refl(int i, int n) {
  return i < 0 ? -i : (i >= n ? 2 * n - 2 - i : i);
}

// ---------------------------------------------------------------------------
// Kernel 1: per-tile 3x3x3 median stencil on d = diff_W(pred_z), accumulating
//   partial sum((d-med)^2) and partial min(d) per workgroup.
// ---------------------------------------------------------------------------
__global__ __launch_bounds__(256) void stencil_kernel(const float* __restrict__ pz,
                                                      float* __restrict__ wsum,
                                                      float* __restrict__ wmin) {
  // (TZ+2) x (TY+2) halo planes, full 193-wide rows, stride padded to 196 (16B multiple)
  __shared__ __align__(16) float sh[TZ + 2][TY + 2][196];
  __shared__ float rsum[256];
  __shared__ float rmin[256];

  const int tid = threadIdx.x;
  const int bz  = (int)blockIdx.x / (H_DIM / TY);
  const int by  = (int)blockIdx.x % (H_DIM / TY);
  const int z0  = bz * TZ, y0 = by * TY;

  // Stage halo tile into LDS with async copies: 60 rows x (48 B128 + 1 B32) chunks.
  for (int t = tid; t < (TZ + 2) * (TY + 2) * 49; t += 256) {
    const int row = t / 49, c = t % 49;
    const int zl = row / (TY + 2), yl = row % (TY + 2);
    const int gz = refl(z0 - 1 + zl, D_DIM);
    const int gy = refl(y0 - 1 + yl, H_DIM);
    const float* grow = pz + ((size_t)gz * H_DIM + gy) * W_DIM;
    if (c < 48) async_copy_b128(lds_off(&sh[zl][yl][c * 4]), grow + c * 4);
    else        async_copy_b32 (lds_off(&sh[zl][yl][192]),   grow + 192);
  }
  wait_async0();
  __syncthreads();

  float acc  = 0.0f;
  float dmin = FBIG;

  for (int i = tid; i < TZ * TY * WD; i += 256) {
    const int x = i % WD;
    const int y = (i / WD) % TY;
    const int z = i / (WD * TY);
    // x-reflect stays inside the 193-wide row
    const int xm = (x > 0)      ? x - 1 : 1;
    const int xp = (x < WD - 1) ? x + 1 : WD - 2;

    // 27 window values + 5 sentinels -> 32-wide Batcher odd-even mergesort.
    float v[32];
#pragma unroll
    for (int oz = 0; oz < 3; ++oz) {
#pragma unroll
      for (int oy = 0; oy < 3; ++oy) {
        const float* row = &sh[z + oz][y + oy][0];
        const int b = (oz * 3 + oy) * 3;
        v[b + 0] = row[xm + 1] - row[xm];
        v[b + 1] = row[x  + 1] - row[x ];
        v[b + 2] = row[xp + 1] - row[xp];
      }
    }
#pragma unroll
    for (int s = 27; s < 32; ++s) v[s] = FBIG;   // sentinels (fold away in CAS)

    const float dc = v[13];           // center d value (oz=1,oy=1,ox=1)
    dmin = fminf(dmin, dc);

    // Batcher odd-even mergesort, n=32: 191 comparators, heavily constant-folded
    // by the 5 sentinel lanes; staged structure gives VOPD dual-issue ILP.
    // (Formulation hand-verified against known comparator counts for n=4 and n=8.)
#pragma unroll
    for (int p = 1; p < 32; p <<= 1) {
#pragma unroll
      for (int k = p; k >= 1; k >>= 1) {
#pragma unroll
        for (int j = (k & (p - 1)); j <= 31 - k; j += 2 * k) {
#pragma unroll
          for (int i2 = 0; i2 <= ((k - 1 < 31 - j - k) ? (k - 1) : (31 - j - k)); ++i2) {
            if ((i2 + j) / (2 * p) == (i2 + j + k) / (2 * p)) {
              const int a = i2 + j, b = i2 + j + k;
              const float lo = fminf(v[a], v[b]);
              const float hi = fmaxf(v[a], v[b]);
              v[a] = lo; v[b] = hi;
            }
          }
        }
      }
    }
    // ascending sort of 27 reals + 5 +MAX sentinels -> lower median at index 13
    const float diff = dc - v[13];
    acc += diff * diff;
  }

  rsum[tid] = acc;
  rmin[tid] = dmin;
  __syncthreads();
  for (int off = 128; off > 0; off >>= 1) {
    if (tid < off) {
      rsum[tid] += rsum[tid + off];
      rmin[tid]  = fminf(rmin[tid], rmin[tid + off]);
    }
    __syncthreads();
  }
  if (tid == 0) {
    wsum[blockIdx.x] = rsum[0];
    wmin[blockIdx.x] = rmin[0];
  }
}

// ---------------------------------------------------------------------------
// Kernel 2: fixed-order finalize of the 768 partials (deterministic across
//   graph replays; no float atomics).
// ---------------------------------------------------------------------------
__global__ __launch_bounds__(256) void finalize_kernel(const float* __restrict__ wsum,
                                                       const float* __restrict__ wmin,
                                                       float* __restrict__ out) {
  __shared__ float ssum[256];
  __shared__ float smin[256];
  const int t = threadIdx.x;
  float s = 0.0f, m = FBIG;
  for (int i = t; i < NBLK; i += 256) { s += wsum[i]; m = fminf(m, wmin[i]); }
  ssum[t] = s; smin[t] = m;
  __syncthreads();
  for (int off = 128; off > 0; off >>= 1) {
    if (t < off) { ssum[t] += ssum[t + off]; smin[t] = fminf(smin[t], smin[t + off]); }
    __syncthreads();
  }
  if (t == 0) {
    out[0] = ssum[0] / (float)NTOT;        // loss_smooth
    out[1] = fmaxf(1.0f - smin[0], 0.0f);  // loss_mon = max(0, max(1 - dz))
  }
}

// ---------------------------------------------------------------------------
// Kernel 3: lower median of z0 = pred_z[..., 0] by exact rank counting.
//   Tiles of 2048 strided-column values are staged to LDS with the TDM
//   (tensor_load_to_lds + hand-packed D#: tile_dim0=1 elem, tile_dim1=2048,
//   tensor_dim0_stride=193), waited with s_wait_tensorcnt.
// ---------------------------------------------------------------------------
__global__ __launch_bounds__(256) void median_z0_kernel(const float* __restrict__ pz,
                                                        float* __restrict__ out) {
  __shared__ __align__(16) float tile[2048];
  const int tid = threadIdx.x;
  const int gid = (int)blockIdx.x * 256 + tid;   // 96 blocks x 256 = 24576 candidates
  const float myv = pz[(size_t)gid * W_DIM];
  int rank = 0;

  for (int base = 0; base < NZ0; base += 2048) {
    if (tid < 32) {  // wave 0 issues the TDM load (duplicates are benign: same data)
      const uint64_t ga = (uint64_t)(uintptr_t)(pz + (size_t)base * W_DIM);
      const u32 tdim1 = (u32)(NZ0 - base);
      u32x4 g0;
      g0.x = 1u;                                             // count=1 valid descriptor
      g0.y = lds_off(&tile[0]);                              // lds_addr
      g0.z = (u32)ga;                                        // global_addr[31:0]
      g0.w = ((u32)(ga >> 32) & 0x01FFFFFFu) | 0x80000000u;  // addr[56:32] | type=2
      u32x8 g1;
      g1.s0 = 0x00020000u;                                   // data_size=4B, no flags
      g1.s1 = (193u & 0xFFFFu) << 16;                        // tensor_dim0 = 193
      g1.s2 = ((193u >> 16) & 0xFFFFu) | ((tdim1 & 0xFFFFu) << 16);  // tensor_dim1
      g1.s3 = ((tdim1 >> 16) & 0xFFFFu) | (1u << 16);        // tile_dim0 = 1
      g1.s4 = 2048u;                                         // tile_dim1 = 2048 rows
      g1.s5 = 193u;                                          // tensor_dim0_stride lo32
      g1.s6 = 0u;                                            // stride hi / dim1_stride
      g1.s7 = 0u;
      asm volatile("tensor_load_to_lds %0, %1" :: "s"(g0), "s"(g1) : "memory");
      __builtin_amdgcn_s_wait_tensorcnt(0);
    }
    __syncthreads();
#pragma unroll 8
    for (int j = 0; j < 2048; ++j) {
      const float o = tile[j];
      const int   oj = base + j;
      rank += ((o < myv) || (o == myv && oj < gid)) ? 1 : 0;
    }
    __syncthreads();
  }

  if (rank == KRANK) out[2] = myv * myv;   // loss_average = median(z0)^2
}

// ---------------------------------------------------------------------------
extern "C" void kernel_launch(void* const* d_in, const int* in_sizes, int n_in,
                              void* d_out, int out_size, void* d_ws, size_t ws_size,
                              hipStream_t stream) {
  (void)in_sizes; (void)n_in; (void)out_size; (void)ws_size;
  const float* pz  = (const float*)d_in[0];
  float* out  = (float*)d_out;
  float* wsum = (float*)d_ws;          // 768 floats
  float* wmin = wsum + NBLK;           // 768 floats

  stencil_kernel  <<<NBLK, 256, 0, stream>>>(pz, wsum, wmin);
  finalize_kernel <<<1,    256, 0, stream>>>(wsum, wmin, out);
  median_z0_kernel<<<96,   256, 0, stream>>>(pz, out);
}